// ModelAttention_55027120996757
// MI455X (gfx1250) — compile-verified
//
#include <hip/hip_runtime.h>
#include <cmath>

// ---------------- problem constants (from reference) ----------------
#define N_NODES   50000
#define N_EDGES   800000
#define F_NODE    128
#define F_EDGE    32
#define HID       32
#define HEADS     4
#define C1        32
#define C2        16
#define N_CLASSES 2

typedef __attribute__((ext_vector_type(2))) float v2f;
typedef __attribute__((ext_vector_type(8))) float v8f;

// ============================================================================
// WMMA f32 GEMM:  Y[nrows, COUT] = X[nrows, K] @ W[K, COUT] (+ bias) (opt relu)
// One wave per 16-row strip; wave holds all CT=ceil(COUT/16) 16-col accumulator
// tiles in VGPRs so A is loaded exactly once per row tile.
// Uses V_WMMA_F32_16X16X4_F32 (keeps full f32 numerics of the reference).
//
// Per ISA layouts (CDNA5 ISA):
//   A 16x4 f32 : lane m=lane&15 holds rows; lane-half h=lane>>4 selects K pair:
//                v0 = A[m][2h], v1 = A[m][2h+1]
//   B 4x16 f32 : lane n=lane&15 holds cols; v0 = B[2h][n], v1 = B[2h+1][n]
//   C/D 16x16  : VGPR j -> row j + 8h, col = lane&15
//
// COUT is compile-time: column masking only exists for COUT=2 (classifier);
// all other instantiations have branch-free unconditional loads/stores.
// ============================================================================
template<int COUT>
__global__ __launch_bounds__(128)
void gemm_wmma_f32(const float* __restrict__ X, const float* __restrict__ W,
                   const float* __restrict__ bias, float* __restrict__ Y,
                   int nrows, int K, int relu)
{
    constexpr int CT   = (COUT + 15) / 16;
    constexpr bool MASK = (COUT % 16) != 0;

    const int lane = threadIdx.x & 31;
    const int wave = threadIdx.x >> 5;
    const int rowTile = blockIdx.x * 4 + wave;
    if (rowTile * 16 >= nrows) return;
    const int row0 = rowTile * 16;
    const int m = lane & 15;
    const int h = lane >> 4;

    v8f acc[CT] = {};
    const float* __restrict__ xrow = X + (size_t)(row0 + m) * K;

    // per-tile B column base pointers (hoisted out of the K loop)
    const float* wcol[CT];
#pragma unroll
    for (int t = 0; t < CT; ++t) wcol[t] = W + t * 16 + m;

#pragma unroll 4
    for (int k0 = 0; k0 < K; k0 += 4) {
        const int ka = k0 + 2 * h;
        v2f a;
        a.x = xrow[ka];
        a.y = xrow[ka + 1];
#pragma unroll
        for (int t = 0; t < CT; ++t) {
            v2f b;
            if (MASK) {
                const bool ok = (t * 16 + m) < COUT;
                b.x = ok ? wcol[t][(size_t)ka * COUT]       : 0.f;
                b.y = ok ? wcol[t][(size_t)(ka + 1) * COUT] : 0.f;
            } else {
                b.x = wcol[t][(size_t)ka * COUT];
                b.y = wcol[t][(size_t)(ka + 1) * COUT];
            }
            acc[t] = __builtin_amdgcn_wmma_f32_16x16x4_f32(
                         /*neg_a=*/false, a, /*neg_b=*/false, b,
                         /*c_mod=*/(short)0, acc[t],
                         /*reuse_a=*/false, /*reuse_b=*/false);
        }
    }

#pragma unroll
    for (int t = 0; t < CT; ++t) {
        const int col = t * 16 + m;
        if (MASK && col >= COUT) continue;
        const float bv = bias ? bias[col] : 0.f;
#pragma unroll
        for (int j = 0; j < 8; ++j) {
            const int row = row0 + j + 8 * h;
            float val = acc[t][j] + bv;
            if (relu) val = fmaxf(val, 0.f);
            Y[(size_t)row * COUT + col] = val;
        }
    }
}

// ---------------- float <-> order-preserving uint encoding (for atomicMax) ----
__device__ __forceinline__ unsigned fenc(float f) {
    unsigned b = __float_as_uint(f);
    return (b & 0x80000000u) ? ~b : (b | 0x80000000u);
}
__device__ __forceinline__ float fdec(unsigned u) {
    unsigned b = (u & 0x80000000u) ? (u ^ 0x80000000u) : ~u;
    return __uint_as_float(b);
}
#define FENC_NEG_INF 0x007FFFFFu   // fenc(-inf)

// ---------------- init fill ----------------
__global__ void fill_u32_kernel(unsigned* __restrict__ p, unsigned v, int n) {
    int i = blockIdx.x * blockDim.x + threadIdx.x;
    if (i < n) p[i] = v;
}

// ---------------- edge pass 1: logits + per-(dst,head) running max ----------
template<int C>
__global__ void edge_logits_max_kernel(const int* __restrict__ src, const int* __restrict__ dst,
                                       const float* __restrict__ q, const float* __restrict__ k,
                                       const float* __restrict__ e, float* __restrict__ logits,
                                       unsigned* __restrict__ mEnc, float inv_sqrt_c)
{
    int t = blockIdx.x * blockDim.x + threadIdx.x;
    if (t >= N_EDGES * HEADS) return;
    const int edge = t >> 2;          // HEADS == 4
    const int head = t & 3;
    const int s = src[edge], d = dst[edge];
    const float4* __restrict__ qv = (const float4*)(q + ((size_t)d * HEADS + head) * C);
    const float4* __restrict__ kv = (const float4*)(k + ((size_t)s * HEADS + head) * C);
    const float4* __restrict__ ev = (const float4*)(e + ((size_t)edge * HEADS + head) * C);
    float acc = 0.f;
#pragma unroll
    for (int i = 0; i < C / 4; ++i) {
        const float4 qq = qv[i], kk = kv[i], ee = ev[i];
        acc += qq.x * (kk.x + ee.x) + qq.y * (kk.y + ee.y)
             + qq.z * (kk.z + ee.z) + qq.w * (kk.w + ee.w);
    }
    const float lg = acc * inv_sqrt_c;
    logits[t] = lg;
    atomicMax(mEnc + (size_t)d * HEADS + head, fenc(lg));
}

// ---------------- edge pass 2: p = exp(logit - max), segment-sum ------------
__global__ void edge_softmax_p_kernel(const int* __restrict__ dst, float* __restrict__ logits,
                                      const unsigned* __restrict__ mEnc, float* __restrict__ ssum)
{
    int t = blockIdx.x * blockDim.x + threadIdx.x;
    if (t >= N_EDGES * HEADS) return;
    const int edge = t >> 2;
    const int head = t & 3;
    const int d = dst[edge];
    const float mval = fdec(mEnc[(size_t)d * HEADS + head]);
    const float p = __expf(logits[t] - mval);
    logits[t] = p;   // overwrite logits with numerator
    __hip_atomic_fetch_add(ssum + (size_t)d * HEADS + head, p,
                           __ATOMIC_RELAXED, __HIP_MEMORY_SCOPE_AGENT);
}

// ---------------- edge pass 3: agg[dst] += alpha * (v[src] + e) -------------
template<int C>
__global__ void edge_aggregate_kernel(const int* __restrict__ src, const int* __restrict__ dst,
                                      const float* __restrict__ p, const float* __restrict__ ssum,
                                      const float* __restrict__ v, const float* __restrict__ e,
                                      float* __restrict__ agg)
{
    constexpr int C4 = C / 4;
    int t = blockIdx.x * blockDim.x + threadIdx.x;
    if (t >= N_EDGES * HEADS * C4) return;
    const int c4   = t % C4;
    const int head = (t / C4) & 3;
    const int edge = t / (C4 * HEADS);
    const int s = src[edge], d = dst[edge];
    const float alpha = p[(size_t)edge * HEADS + head] / ssum[(size_t)d * HEADS + head];
    const float4 vv = *(const float4*)(v + ((size_t)s    * HEADS + head) * C + c4 * 4);
    const float4 ee = *(const float4*)(e + ((size_t)edge * HEADS + head) * C + c4 * 4);
    float* __restrict__ out = agg + ((size_t)d * HEADS + head) * C + c4 * 4;
    __hip_atomic_fetch_add(out + 0, alpha * (vv.x + ee.x), __ATOMIC_RELAXED, __HIP_MEMORY_SCOPE_AGENT);
    __hip_atomic_fetch_add(out + 1, alpha * (vv.y + ee.y), __ATOMIC_RELAXED, __HIP_MEMORY_SCOPE_AGENT);
    __hip_atomic_fetch_add(out + 2, alpha * (vv.z + ee.z), __ATOMIC_RELAXED, __HIP_MEMORY_SCOPE_AGENT);
    __hip_atomic_fetch_add(out + 3, alpha * (vv.w + ee.w), __ATOMIC_RELAXED, __HIP_MEMORY_SCOPE_AGENT);
}

// ---------------- node combine: relu(mean_heads(agg) + skip) ----------------
template<int C>
__global__ void combine_kernel(const float* __restrict__ agg, const float* __restrict__ skip,
                               float* __restrict__ out)
{
    int t = blockIdx.x * blockDim.x + threadIdx.x;
    if (t >= N_NODES * C) return;
    const int node = t / C;
    const int c = t % C;
    float s = 0.f;
#pragma unroll
    for (int hh = 0; hh < HEADS; ++hh) s += agg[((size_t)node * HEADS + hh) * C + c];
    out[t] = fmaxf(s * (1.f / HEADS) + skip[t], 0.f);
}

// ============================================================================
// Host-side orchestration
// ============================================================================
static inline int cdiv(int a, int b) { return (a + b - 1) / b; }

static void launch_gemm(const float* X, const float* W, const float* bias, float* Y,
                        int nrows, int K, int cout, int relu, hipStream_t stream)
{
    const int rowTiles = cdiv(nrows, 16);
    dim3 block(128);
    dim3 grid(cdiv(rowTiles, 4));
    switch (cout) {
        case 128: gemm_wmma_f32<128><<<grid, block, 0, stream>>>(X, W, bias, Y, nrows, K, relu); break;
        case 64:  gemm_wmma_f32<64> <<<grid, block, 0, stream>>>(X, W, bias, Y, nrows, K, relu); break;
        case 32:  gemm_wmma_f32<32> <<<grid, block, 0, stream>>>(X, W, bias, Y, nrows, K, relu); break;
        case 16:  gemm_wmma_f32<16> <<<grid, block, 0, stream>>>(X, W, bias, Y, nrows, K, relu); break;
        default:  gemm_wmma_f32<2>  <<<grid, block, 0, stream>>>(X, W, bias, Y, nrows, K, relu); break;
    }
}

template<int C>
static void run_edge_phase(const int* src, const int* dst, const float* q, const float* k,
                           const float* v, const float* e, float* logits, unsigned* mEnc,
                           float* ssum, float* agg, hipStream_t stream)
{
    const int NH = N_NODES * HEADS;
    fill_u32_kernel<<<cdiv(NH, 256), 256, 0, stream>>>(mEnc, FENC_NEG_INF, NH);
    fill_u32_kernel<<<cdiv(NH, 256), 256, 0, stream>>>((unsigned*)ssum, 0u, NH);
    const int aggN = N_NODES * HEADS * C;
    fill_u32_kernel<<<cdiv(aggN, 256), 256, 0, stream>>>((unsigned*)agg, 0u, aggN);

    const int EH = N_EDGES * HEADS;
    const float inv_sqrt_c = 1.0f / sqrtf((float)C);
    edge_logits_max_kernel<C><<<cdiv(EH, 256), 256, 0, stream>>>(src, dst, q, k, e, logits, mEnc, inv_sqrt_c);
    edge_softmax_p_kernel<<<cdiv(EH, 256), 256, 0, stream>>>(dst, logits, mEnc, ssum);
    const int tot = N_EDGES * HEADS * (C / 4);
    edge_aggregate_kernel<C><<<cdiv(tot, 256), 256, 0, stream>>>(src, dst, logits, ssum, v, e, agg);
}

extern "C" void kernel_launch(void* const* d_in, const int* in_sizes, int n_in,
                              void* d_out, int out_size, void* d_ws, size_t ws_size,
                              hipStream_t stream)
{
    (void)in_sizes; (void)n_in; (void)out_size; (void)ws_size;

    const float* x    = (const float*)d_in[0];
    const int*   eidx = (const int*)d_in[1];
    const float* ea   = (const float*)d_in[2];
    const float *Wq1 = (const float*)d_in[3],  *bq1 = (const float*)d_in[4];
    const float *Wk1 = (const float*)d_in[5],  *bk1 = (const float*)d_in[6];
    const float *Wv1 = (const float*)d_in[7],  *bv1 = (const float*)d_in[8];
    const float *We1 = (const float*)d_in[9];
    const float *Ws1 = (const float*)d_in[10], *bs1 = (const float*)d_in[11];
    const float *Wq2 = (const float*)d_in[12], *bq2 = (const float*)d_in[13];
    const float *Wk2 = (const float*)d_in[14], *bk2 = (const float*)d_in[15];
    const float *Wv2 = (const float*)d_in[16], *bv2 = (const float*)d_in[17];
    const float *We2 = (const float*)d_in[18];
    const float *Ws2 = (const float*)d_in[19], *bs2 = (const float*)d_in[20];
    const float *W3  = (const float*)d_in[21], *b3  = (const float*)d_in[22];
    const float *W4  = (const float*)d_in[23], *b4  = (const float*)d_in[24];

    const int* src = eidx;            // edge_index[0]
    const int* dst = eidx + N_EDGES;  // edge_index[1]

    // ---- workspace carve (regions reused between conv1 and conv2) ----
    size_t off = 0;
    auto carve = [&](size_t bytes) -> char* {
        char* p = (char*)d_ws + off;
        off += (bytes + 255) & ~(size_t)255;
        return p;
    };
    float*    q      = (float*)carve((size_t)N_NODES * 128 * 4);   // q1 / q2
    float*    k      = (float*)carve((size_t)N_NODES * 128 * 4);   // k1 / k2
    float*    v      = (float*)carve((size_t)N_NODES * 128 * 4);   // v1 / v2
    float*    e      = (float*)carve((size_t)N_EDGES * 128 * 4);   // e1 / e2 (biggest: 410MB)
    float*    logits = (float*)carve((size_t)N_EDGES * HEADS * 4); // logits -> p
    unsigned* mEnc   = (unsigned*)carve((size_t)N_NODES * HEADS * 4);
    float*    ssum   = (float*)carve((size_t)N_NODES * HEADS * 4);
    float*    agg    = (float*)carve((size_t)N_NODES * 128 * 4);   // agg1 / agg2
    float*    skip   = (float*)carve((size_t)N_NODES * C1 * 4);    // skip1 / skip2
    float*    h1     = (float*)carve((size_t)N_NODES * C1 * 4);
    float*    h2     = (float*)carve((size_t)N_NODES * C2 * 4);
    float*    h3     = (float*)carve((size_t)N_NODES * HID * 4);

    // ================= conv1 =================
    launch_gemm(x,  Wq1, bq1,     q,    N_NODES, F_NODE, HEADS * C1, 0, stream);
    launch_gemm(x,  Wk1, bk1,     k,    N_NODES, F_NODE, HEADS * C1, 0, stream);
    launch_gemm(x,  Wv1, bv1,     v,    N_NODES, F_NODE, HEADS * C1, 0, stream);
    launch_gemm(ea, We1, nullptr, e,    N_EDGES, F_EDGE, HEADS * C1, 0, stream);
    launch_gemm(x,  Ws1, bs1,     skip, N_NODES, F_NODE, C1,         0, stream);
    run_edge_phase<C1>(src, dst, q, k, v, e, logits, mEnc, ssum, agg, stream);
    combine_kernel<C1><<<cdiv(N_NODES * C1, 256), 256, 0, stream>>>(agg, skip, h1);

    // ================= conv2 =================
    launch_gemm(h1, Wq2, bq2,     q,    N_NODES, C1,     HEADS * C2, 0, stream);
    launch_gemm(h1, Wk2, bk2,     k,    N_NODES, C1,     HEADS * C2, 0, stream);
    launch_gemm(h1, Wv2, bv2,     v,    N_NODES, C1,     HEADS * C2, 0, stream);
    launch_gemm(ea, We2, nullptr, e,    N_EDGES, F_EDGE, HEADS * C2, 0, stream);
    launch_gemm(h1, Ws2, bs2,     skip, N_NODES, C1,     C2,         0, stream);
    run_edge_phase<C2>(src, dst, q, k, v, e, logits, mEnc, ssum, agg, stream);
    combine_kernel<C2><<<cdiv(N_NODES * C2, 256), 256, 0, stream>>>(agg, skip, h2);

    // ================= dense head =================
    launch_gemm(h2, W3, b3, h3,             N_NODES, C2,  HID,       1, stream);
    launch_gemm(h3, W4, b4, (float*)d_out,  N_NODES, HID, N_CLASSES, 0, stream);
}